// GRDPINet_86870008529010
// MI455X (gfx1250) — compile-verified
//
#include <hip/hip_runtime.h>
#include <cfloat>
#include <cstdint>

// Problem geometry (fixed by the reference).
#define TLEN   2048      // time steps per row
#define BROWS  8192      // batch rows
#define NCH    16        // chunks per row
#define CHUNK  128       // time steps per chunk (NCH*CHUNK == TLEN)
#define SUBC   16        // columns per LDS sub-tile
#define RPB    256       // rows per block (== blockDim.x)
#define LSTRIDE 20       // LDS row stride in floats (16 data + 4 pad; 80B = 16B aligned, 2-way banks)
#define EPSF   1e-10f

// ---- CDNA5 async LDS<->global data movers (ASYNCcnt path, CDNA5 ISA) ----
// GVS mode: 64-bit SGPR base + 32-bit per-lane byte offset; vdst/vsrc hold LDS byte address.
// Temporal hints (CDNA5 ISA) manage 192MB-L2 residency across passes.
__device__ __forceinline__ void async_ld_b128_ht(uint32_t lds_addr, const void* sbase, uint32_t voff) {
  // Pass-1 loads: high-priority temporal -> keep x/x_pre resident in L2 for the pass-3 re-read.
  asm volatile("global_load_async_to_lds_b128 %0, %1, %2 th:TH_LOAD_HT"
               :: "v"(lds_addr), "v"(voff), "s"(sbase) : "memory");
}
__device__ __forceinline__ void async_ld_b128_lu(uint32_t lds_addr, const void* sbase, uint32_t voff) {
  // Pass-3 loads: last-use -> drop the line after this read, freeing L2 as we go.
  asm volatile("global_load_async_to_lds_b128 %0, %1, %2 th:TH_LOAD_LU"
               :: "v"(lds_addr), "v"(voff), "s"(sbase) : "memory");
}
__device__ __forceinline__ void async_st_b128_nt(uint32_t voff, uint32_t lds_addr, void* sbase) {
  // Output stores: non-temporal -> 64MB of never-re-read data must not evict x/x_pre from L2.
  asm volatile("global_store_async_from_lds_b128 %0, %1, %2 th:TH_STORE_NT"
               :: "v"(voff), "v"(lds_addr), "s"(sbase) : "memory");
}
__device__ __forceinline__ void wait_async0() { asm volatile("s_wait_asynccnt 0" ::: "memory"); }
__device__ __forceinline__ void wait_ds0()    { asm volatile("s_wait_dscnt 0"    ::: "memory"); }

// Pass 1: per (row, chunk) compute the composed clamp (L[4], H[4]) of the chunk.
// z_t = clamp(z_{t-1}, u-r, u+r); composition: L' = clamp(L, lo, hi), H' = clamp(H, lo, hi).
__global__ void __launch_bounds__(RPB) pass_clamps(const float* __restrict__ x,
                                                   const float* __restrict__ xp,
                                                   const float* __restrict__ w,
                                                   float* __restrict__ clamps) {
  __shared__ float smem[RPB * LSTRIDE * 2];   // x tile | x_pre tile (40 KB)
  const int tid    = threadIdx.x;
  const int rowBlk = blockIdx.x & 31;         // BROWS / RPB = 32
  const int chunk  = blockIdx.x >> 5;
  const int rowBase = rowBlk * RPB;
  const int myRow   = rowBase + tid;

  const float h = w[5], beta = w[6];
  const float hI0 = h * 1.f, hI1 = h * 2.f, hI2 = h * 3.f, hI3 = h * 4.f;

  float L[4], H[4];
#pragma unroll
  for (int i = 0; i < 4; ++i) { L[i] = -FLT_MAX; H[i] = FLT_MAX; }

  // Wave-local async staging: wave w loads rows [w*32, w*32+32); its lanes consume only those,
  // so s_wait_asynccnt alone suffices (no workgroup barriers).
  const int lane = tid & 31, wv = tid >> 5;
  const int rIn  = lane >> 2;        // 0..7
  const int seg  = lane & 3;         // 0..3 (16B segments)
  const uint32_t sx  = (uint32_t)(uintptr_t)&smem[0];
  const uint32_t sxp = sx + RPB * LSTRIDE * 4;

  for (int sub = 0; sub < CHUNK / SUBC; ++sub) {
    const int t0 = chunk * CHUNK + sub * SUBC;
#pragma unroll
    for (int j = 0; j < 4; ++j) {
      const int r = wv * 32 + j * 8 + rIn;
      const uint32_t goff = (uint32_t)(((rowBase + r) * TLEN + t0 + seg * 4) * 4);
      const uint32_t loff = (uint32_t)((r * LSTRIDE + seg * 4) * 4);
      async_ld_b128_ht(sx + loff, x, goff);
      async_ld_b128_ht(sxp + loff, xp, goff);
    }
    wait_async0();
#pragma unroll
    for (int tl = 0; tl < SUBC; ++tl) {
      const float u  = smem[tid * LSTRIDE + tl];
      const float up = smem[RPB * LSTRIDE + tid * LSTRIDE + tl];
      const float du = u - up;
      const float s  = __builtin_sqrtf(__builtin_fmaf(du, du, EPSF));
      const float bs = beta * s;
      const float rr[4] = {hI0 + bs, hI1 + bs, hI2 + bs, hI3 + bs};
#pragma unroll
      for (int i = 0; i < 4; ++i) {
        const float lo = u - rr[i], hi = u + rr[i];
        H[i] = fminf(fmaxf(H[i], lo), hi);
        L[i] = fminf(fmaxf(L[i], lo), hi);
      }
    }
  }
  float4* cp = (float4*)clamps;
  const int base = (chunk * BROWS + myRow) * 2;
  cp[base + 0] = make_float4(L[0], L[1], L[2], L[3]);
  cp[base + 1] = make_float4(H[0], H[1], H[2], H[3]);
}

// Pass 2: stitch chunk boundaries sequentially (16 steps), record chunk-entry states,
// emit z_final and the weights passthrough into d_out.
__global__ void __launch_bounds__(RPB) pass_stitch(const float* __restrict__ z_init,
                                                   const float* __restrict__ w,
                                                   const float* __restrict__ clamps,
                                                   float* __restrict__ zstart,
                                                   float* __restrict__ out) {
  const int r = blockIdx.x * RPB + threadIdx.x;   // 32 blocks x 256
  float4 z = ((const float4*)z_init)[r];
  float zc[4] = {z.x, z.y, z.z, z.w};
  for (int c = 0; c < NCH; ++c) {
    ((float4*)zstart)[c * BROWS + r] = make_float4(zc[0], zc[1], zc[2], zc[3]);
    const float4 Lv = ((const float4*)clamps)[(c * BROWS + r) * 2];
    const float4 Hv = ((const float4*)clamps)[(c * BROWS + r) * 2 + 1];
    zc[0] = fminf(fmaxf(zc[0], Lv.x), Hv.x);
    zc[1] = fminf(fmaxf(zc[1], Lv.y), Hv.y);
    zc[2] = fminf(fmaxf(zc[2], Lv.z), Hv.z);
    zc[3] = fminf(fmaxf(zc[3], Lv.w), Hv.w);
  }
  const long long zoff = (long long)BROWS * TLEN + 11;
#pragma unroll
  for (int i = 0; i < 4; ++i) out[zoff + (long long)r * 4 + i] = zc[i];
  if (blockIdx.x == 0 && threadIdx.x < 11)
    out[(long long)BROWS * TLEN + threadIdx.x] = w[threadIdx.x];
}

// Pass 3: replay each chunk with its known entry state; outputs staged in LDS and
// drained with coalesced non-temporal async stores.
__global__ void __launch_bounds__(RPB) pass_emit(const float* __restrict__ x,
                                                 const float* __restrict__ xp,
                                                 const float* __restrict__ w,
                                                 const float* __restrict__ zstart,
                                                 float* __restrict__ out) {
  __shared__ float smem[RPB * LSTRIDE * 3];   // x | x_pre | out tiles (60 KB)
  const int tid    = threadIdx.x;
  const int rowBlk = blockIdx.x & 31;
  const int chunk  = blockIdx.x >> 5;
  const int rowBase = rowBlk * RPB;
  const int myRow   = rowBase + tid;

  const float a0 = w[0], a1 = w[1], a2 = w[2], a3 = w[3], a4 = w[4];
  const float h = w[5], beta = w[6], bb = w[7];
  const float hI0 = h * 1.f, hI1 = h * 2.f, hI2 = h * 3.f, hI3 = h * 4.f;

  const float4 z0 = ((const float4*)zstart)[chunk * BROWS + myRow];
  float zc[4] = {z0.x, z0.y, z0.z, z0.w};

  const int lane = tid & 31, wv = tid >> 5;
  const int rIn  = lane >> 2;
  const int seg  = lane & 3;
  const uint32_t sx   = (uint32_t)(uintptr_t)&smem[0];
  const uint32_t sxp  = sx + RPB * LSTRIDE * 4;
  const uint32_t sout = sx + RPB * LSTRIDE * 8;

  for (int sub = 0; sub < CHUNK / SUBC; ++sub) {
    const int t0 = chunk * CHUNK + sub * SUBC;
#pragma unroll
    for (int j = 0; j < 4; ++j) {
      const int r = wv * 32 + j * 8 + rIn;
      const uint32_t goff = (uint32_t)(((rowBase + r) * TLEN + t0 + seg * 4) * 4);
      const uint32_t loff = (uint32_t)((r * LSTRIDE + seg * 4) * 4);
      async_ld_b128_lu(sx + loff, x, goff);
      async_ld_b128_lu(sxp + loff, xp, goff);
    }
    wait_async0();   // also drains previous sub-tile's async stores before we rewrite the out tile
#pragma unroll
    for (int tl = 0; tl < SUBC; ++tl) {
      const float u  = smem[tid * LSTRIDE + tl];
      const float up = smem[RPB * LSTRIDE + tid * LSTRIDE + tl];
      const float du = u - up;
      const float s  = __builtin_sqrtf(__builtin_fmaf(du, du, EPSF));
      const float bs = beta * s;
      const float rr[4] = {hI0 + bs, hI1 + bs, hI2 + bs, hI3 + bs};
#pragma unroll
      for (int i = 0; i < 4; ++i) {
        const float lo = u - rr[i], hi = u + rr[i];
        zc[i] = fminf(fmaxf(zc[i], lo), hi);
      }
      float o = __builtin_fmaf(a0, u, bb);
      o = __builtin_fmaf(zc[0], a1, o);
      o = __builtin_fmaf(zc[1], a2, o);
      o = __builtin_fmaf(zc[2], a3, o);
      o = __builtin_fmaf(zc[3], a4, o);
      smem[RPB * LSTRIDE * 2 + tid * LSTRIDE + tl] = o;
    }
    wait_ds0();      // LDS out-tile writes visible before async DMA reads them
#pragma unroll
    for (int j = 0; j < 4; ++j) {
      const int r = wv * 32 + j * 8 + rIn;
      const uint32_t goff = (uint32_t)(((rowBase + r) * TLEN + t0 + seg * 4) * 4);
      const uint32_t loff = (uint32_t)((r * LSTRIDE + seg * 4) * 4);
      async_st_b128_nt(goff, sout + loff, out);
    }
  }
  // s_endpgm performs an implicit wait-idle, draining the final async stores.
}

extern "C" void kernel_launch(void* const* d_in, const int* in_sizes, int n_in,
                              void* d_out, int out_size, void* d_ws, size_t ws_size,
                              hipStream_t stream) {
  const float* x      = (const float*)d_in[0];   // (8192, 2048, 1) f32
  const float* x_pre  = (const float*)d_in[1];   // (8192, 2048, 1) f32
  const float* z_init = (const float*)d_in[2];   // (8192, 1, 4)   f32
  const float* wts    = (const float*)d_in[3];   // (11,)          f32
  float* out = (float*)d_out;                    // [B*T | 11 | B*4] f32

  float* clamps = (float*)d_ws;                               // NCH*BROWS*8 floats = 4 MB
  float* zstart = (float*)d_ws + (size_t)NCH * BROWS * 8;     // NCH*BROWS*4 floats = 2 MB

  pass_clamps<<<(BROWS / RPB) * NCH, RPB, 0, stream>>>(x, x_pre, wts, clamps);
  pass_stitch<<<BROWS / RPB, RPB, 0, stream>>>(z_init, wts, clamps, zstart, out);
  pass_emit  <<<(BROWS / RPB) * NCH, RPB, 0, stream>>>(x, x_pre, wts, zstart, out);
}